// SemanticAttention_30142080483330
// MI455X (gfx1250) — compile-verified
//
#include <hip/hip_runtime.h>
#include <hip/hip_bf16.h>

typedef __attribute__((ext_vector_type(2))) float v2f;
typedef __attribute__((ext_vector_type(8))) float v8f;

#define FDIM 128
#define MPATH 4
#define BIGIDX 0x7fffffff

// Branch-free tanh: tanh(x) = 1 - 2*rcp(exp(2x)+1).
// __expf -> v_exp_f32 (TRANS), __builtin_amdgcn_rcpf -> single v_rcp_f32.
__device__ __forceinline__ float tanh_fast(float x) {
    float e = __expf(2.0f * x);
    return fmaf(-2.0f, __builtin_amdgcn_rcpf(e + 1.0f), 1.0f);
}

// ---------------------------------------------------------------------------
// K0: init first-occurrence table
__global__ void init_fiv_kernel(int* __restrict__ fiv, int n) {
    int v = blockIdx.x * 256 + threadIdx.x;
    if (v < n) fiv[v] = BIGIDX;
}

// K1: first occurrence of each value (values bounded by L -> counting, no sort)
__global__ void firstocc_kernel(const int* __restrict__ nodes, int* __restrict__ fiv, int n) {
    int i = blockIdx.x * 256 + threadIdx.x;
    if (i < n) atomicMin(&fiv[nodes[i]], i);
}

// K2a: per-block exclusive scan of presence flags
__global__ void scan_block_kernel(const int* __restrict__ fiv, int* __restrict__ partial,
                                  int* __restrict__ blockSums, int n) {
    __shared__ int sm[256];
    int t = threadIdx.x;
    int gid = blockIdx.x * 256 + t;
    int flag = (gid < n && fiv[gid] != BIGIDX) ? 1 : 0;
    int val = flag;
    sm[t] = val;
    __syncthreads();
    for (int off = 1; off < 256; off <<= 1) {
        int add = (t >= off) ? sm[t - off] : 0;
        __syncthreads();
        val += add;
        sm[t] = val;
        __syncthreads();
    }
    if (gid < n) partial[gid] = val - flag;      // exclusive
    if (t == 255) blockSums[blockIdx.x] = val;   // block total
}

// K2b: scan of block sums (single block; nb <= 512 since L <= 131072)
__global__ void scan_sums_kernel(const int* __restrict__ blockSums, int* __restrict__ blockOffsets,
                                 int* __restrict__ numUnique, int nb) {
    __shared__ int sm[512];
    int t = threadIdx.x;
    int v = (t < nb) ? blockSums[t] : 0;
    int val = v;
    sm[t] = val;
    __syncthreads();
    for (int off = 1; off < 512; off <<= 1) {
        int add = (t >= off) ? sm[t - off] : 0;
        __syncthreads();
        val += add;
        sm[t] = val;
        __syncthreads();
    }
    if (t < nb) blockOffsets[t] = val - v;
    if (t == 511) *numUnique = val;
}

// K3: rank[v] = #present values < v  (== searchsorted(uniq, v)); scatter first-idx by rank
__global__ void build_sorted_kernel(const int* __restrict__ fiv, const int* __restrict__ partial,
                                    const int* __restrict__ blockOffsets, int* __restrict__ rank,
                                    int* __restrict__ fiSorted, int n) {
    int v = blockIdx.x * 256 + threadIdx.x;
    if (v >= n) return;
    int fi = fiv[v];
    int r = partial[v] + blockOffsets[blockIdx.x];
    rank[v] = r;
    if (fi != BIGIDX) fiSorted[r] = fi;
}

// K4: pad first-idx beyond numUnique with fiSorted[0] (matches JAX nonzero(size=) zero-pad -> perm[0])
__global__ void pad_sorted_kernel(int* __restrict__ fiSorted, const int* __restrict__ numUnique, int n) {
    int r = blockIdx.x * 256 + threadIdx.x;
    if (r >= n) return;
    int nu = *numUnique;
    if (r >= nu) fiSorted[r] = fiSorted[0];
}

// ---------------------------------------------------------------------------
// Stage 1: H[m,j,:] = tanh(nodeF[m, fiSorted[j], :] @ trans + bias)
// One wave per 16-node tile; f32 WMMA 16x16x4, K=128.
__global__ __launch_bounds__(64) void stage1_kernel(
    const float* __restrict__ nodeF, const int* __restrict__ fiSorted,
    const float* __restrict__ trans, const float* __restrict__ bias,
    float* __restrict__ H, int Lc)
{
    __shared__ __align__(16) float smem[2][16 * FDIM];
    const int lane = threadIdx.x & 31;
    const int wave = __builtin_amdgcn_readfirstlane(threadIdx.x >> 5);
    const int m = blockIdx.y;
    const int tile = blockIdx.x * 2 + wave;
    const int j0 = tile * 16;
    if (j0 >= Lc) return;
    const bool full = (j0 + 16 <= Lc);   // wave-uniform (always true when L%16==0)
    float* sm = smem[wave];
    const size_t mbase = (size_t)m * Lc * FDIM;

    // gather 16 feature rows into LDS (each row 512B contiguous); row index clamped
    for (int r = 0; r < 16; ++r) {
        int jr = j0 + r; if (jr >= Lc) jr = Lc - 1;
        int src = fiSorted[jr];
        const float4* p = (const float4*)(nodeF + mbase + (size_t)src * FDIM);
        ((float4*)(sm + r * FDIM))[lane] = p[lane];
    }
    __builtin_amdgcn_wave_barrier();

    const int row   = lane & 15;            // A-matrix row (node within tile)
    const int khalf = (lane >> 4) << 1;     // 0 or 2: K sub-lane offset
    for (int nt = 0; nt < 8; ++nt) {
        const int col = nt * 16 + (lane & 15);
        v8f acc = {};
        #pragma unroll 4
        for (int kc = 0; kc < 32; ++kc) {
            const int kb = kc * 4 + khalf;
            v2f a = *(const v2f*)(sm + row * FDIM + kb);   // A[row][kb], A[row][kb+1]
            v2f b;
            b.x = trans[(size_t)kb * FDIM + col];          // B[kb][col]
            b.y = trans[(size_t)(kb + 1) * FDIM + col];    // B[kb+1][col]
            acc = __builtin_amdgcn_wmma_f32_16x16x4_f32(
                false, a, false, b, (short)0, acc, false, false);
        }
        const float bc = bias[col];
        const int rbase = (lane >> 4) * 8;
        float* hp = H + mbase + (size_t)(j0 + rbase) * FDIM + col;
        if (full) {                          // uniform branch: no per-lane EXEC churn
            #pragma unroll
            for (int i = 0; i < 8; ++i)
                hp[(size_t)i * FDIM] = tanh_fast(acc[i] + bc);
        } else {
            #pragma unroll
            for (int i = 0; i < 8; ++i)
                if (j0 + rbase + i < Lc)
                    hp[(size_t)i * FDIM] = tanh_fast(acc[i] + bc);
        }
    }
}

// ---------------------------------------------------------------------------
// Stage 2: S[m,j] = sum_f tanh(H[m,j,:] @ W + b)[f] * q[f]
__global__ __launch_bounds__(64) void stage2_kernel(
    const float* __restrict__ H, const float* __restrict__ W,
    const float* __restrict__ bvec, const float* __restrict__ q,
    float* __restrict__ S, int Lc)
{
    __shared__ __align__(16) float smem[2][16 * FDIM];
    const int lane = threadIdx.x & 31;
    const int wave = __builtin_amdgcn_readfirstlane(threadIdx.x >> 5);
    const int m = blockIdx.y;
    const int tile = blockIdx.x * 2 + wave;
    const int j0 = tile * 16;
    if (j0 >= Lc) return;
    const bool full = (j0 + 16 <= Lc);
    float* sm = smem[wave];
    const size_t mbase = (size_t)m * Lc * FDIM;

    // stage 16x128 H tile into LDS; row index clamped
    for (int r = 0; r < 16; ++r) {
        int jr = j0 + r; if (jr >= Lc) jr = Lc - 1;
        const float4* p = (const float4*)(H + mbase + (size_t)jr * FDIM);
        ((float4*)(sm + r * FDIM))[lane] = p[lane];
    }
    __builtin_amdgcn_wave_barrier();

    const int row   = lane & 15;
    const int khalf = (lane >> 4) << 1;
    float sdot[8];
    #pragma unroll
    for (int i = 0; i < 8; ++i) sdot[i] = 0.0f;

    for (int nt = 0; nt < 8; ++nt) {
        const int col = nt * 16 + (lane & 15);
        v8f acc = {};
        #pragma unroll 4
        for (int kc = 0; kc < 32; ++kc) {
            const int kb = kc * 4 + khalf;
            v2f a = *(const v2f*)(sm + row * FDIM + kb);
            v2f b;
            b.x = W[(size_t)kb * FDIM + col];
            b.y = W[(size_t)(kb + 1) * FDIM + col];
            acc = __builtin_amdgcn_wmma_f32_16x16x4_f32(
                false, a, false, b, (short)0, acc, false, false);
        }
        const float bc = bvec[col];
        const float qc = q[col];
        #pragma unroll
        for (int i = 0; i < 8; ++i)
            sdot[i] += tanh_fast(acc[i] + bc) * qc;   // partial dot over this lane's column
    }
    // reduce across the 16 lanes of each half (cols are disjoint per lane)
    #pragma unroll
    for (int i = 0; i < 8; ++i) {
        float v = sdot[i];
        v += __shfl_xor(v, 1, 32);
        v += __shfl_xor(v, 2, 32);
        v += __shfl_xor(v, 4, 32);
        v += __shfl_xor(v, 8, 32);
        sdot[i] = v;
    }
    if ((lane & 15) == 0) {
        const int rbase = (lane >> 4) * 8;
        if (full) {
            #pragma unroll
            for (int i = 0; i < 8; ++i)
                S[(size_t)m * Lc + j0 + rbase + i] = sdot[i];
        } else {
            #pragma unroll
            for (int i = 0; i < 8; ++i)
                if (j0 + rbase + i < Lc) S[(size_t)m * Lc + j0 + rbase + i] = sdot[i];
        }
    }
}

// ---------------------------------------------------------------------------
// K7: beta = softmax_p( (S0 + S_{p+1})/2 );  also emit all_weight = beta.T
__global__ void beta_kernel(const float* __restrict__ S, float* __restrict__ betaW,
                            float* __restrict__ out_allw, int Lc) {
    int j = blockIdx.x * 256 + threadIdx.x;
    if (j >= Lc) return;
    float s0 = S[j];
    float w0 = 0.5f * (s0 + S[(size_t)Lc + j]);
    float w1 = 0.5f * (s0 + S[(size_t)2 * Lc + j]);
    float w2 = 0.5f * (s0 + S[(size_t)3 * Lc + j]);
    float mx = fmaxf(w0, fmaxf(w1, w2));
    float e0 = __expf(w0 - mx), e1 = __expf(w1 - mx), e2 = __expf(w2 - mx);
    float inv = __builtin_amdgcn_rcpf(e0 + e1 + e2);
    float b0 = e0 * inv, b1 = e1 * inv, b2 = e2 * inv;
    betaW[3 * j + 0] = b0;  betaW[3 * j + 1] = b1;  betaW[3 * j + 2] = b2;
    out_allw[3 * j + 0] = b0; out_allw[3 * j + 1] = b1; out_allw[3 * j + 2] = b2;
}

// K8: result[i,f] = h0[j,f] + sum_p beta[p,j] * h_{p+1}[j,f],  j = rank[nodes_ori[i]]
__global__ __launch_bounds__(128) void result_kernel(
    const float* __restrict__ H, const float* __restrict__ betaW,
    const int* __restrict__ rank, const int* __restrict__ nodes,
    float* __restrict__ out, int Lc) {
    int i = blockIdx.x;
    int f = threadIdx.x;
    int j = rank[nodes[i]];
    float b0 = betaW[3 * j + 0], b1 = betaW[3 * j + 1], b2 = betaW[3 * j + 2];
    const float* h = H + (size_t)j * FDIM;
    size_t stride = (size_t)Lc * FDIM;
    float x = h[f] + b0 * h[stride + f] + b1 * h[2 * stride + f] + b2 * h[3 * stride + f];
    out[(size_t)i * FDIM + f] = x;
}

// ---------------------------------------------------------------------------
extern "C" void kernel_launch(void* const* d_in, const int* in_sizes, int n_in,
                              void* d_out, int out_size, void* d_ws, size_t ws_size,
                              hipStream_t stream) {
    const float* nodeF  = (const float*)d_in[0];   // (M, L, F)
    const int*   nodes  = (const int*)  d_in[1];   // (L,)
    const float* trans  = (const float*)d_in[2];   // (F, F)
    const float* W      = (const float*)d_in[3];   // (F, F)
    const float* bvec   = (const float*)d_in[4];   // (1, F)
    const float* q      = (const float*)d_in[5];   // (F, 1)
    const float* bias   = (const float*)d_in[6];   // (F,)
    const int Lc = in_sizes[1];
    (void)n_in; (void)out_size; (void)ws_size;

    char* ws = (char*)d_ws;
    size_t off = 0;
    auto take = [&](size_t bytes) -> void* {
        void* p = (void*)(ws + off);
        off += (bytes + 255) & ~(size_t)255;
        return p;
    };
    int*   fiv          = (int*)  take((size_t)Lc * sizeof(int));
    int*   partial      = (int*)  take((size_t)Lc * sizeof(int));
    int*   blockSums    = (int*)  take(512 * sizeof(int));
    int*   blockOffsets = (int*)  take(512 * sizeof(int));
    int*   numUnique    = (int*)  take(256);
    int*   rank         = (int*)  take((size_t)Lc * sizeof(int));
    int*   fiSorted     = (int*)  take((size_t)Lc * sizeof(int));
    float* betaW        = (float*)take((size_t)3 * Lc * sizeof(float));
    float* S            = (float*)take((size_t)MPATH * Lc * sizeof(float));
    float* H            = (float*)take((size_t)MPATH * Lc * FDIM * sizeof(float));

    const int nb = (Lc + 255) / 256;

    init_fiv_kernel   <<<nb, 256, 0, stream>>>(fiv, Lc);
    firstocc_kernel   <<<nb, 256, 0, stream>>>(nodes, fiv, Lc);
    scan_block_kernel <<<nb, 256, 0, stream>>>(fiv, partial, blockSums, Lc);
    scan_sums_kernel  <<<1, 512, 0, stream>>>(blockSums, blockOffsets, numUnique, nb);
    build_sorted_kernel<<<nb, 256, 0, stream>>>(fiv, partial, blockOffsets, rank, fiSorted, Lc);
    pad_sorted_kernel <<<nb, 256, 0, stream>>>(fiSorted, numUnique, Lc);

    const int tiles = (Lc + 15) / 16;
    dim3 gs((tiles + 1) / 2, MPATH);
    stage1_kernel<<<gs, 64, 0, stream>>>(nodeF, fiSorted, trans, bias, H, Lc);
    stage2_kernel<<<gs, 64, 0, stream>>>(H, W, bvec, q, S, Lc);

    float* out_result = (float*)d_out;
    float* out_allw   = out_result + (size_t)Lc * FDIM;
    beta_kernel  <<<nb, 256, 0, stream>>>(S, betaW, out_allw, Lc);
    result_kernel<<<Lc, 128, 0, stream>>>(H, betaW, rank, nodes, out_result, Lc);
}